// PPIN_DC_20753281974843
// MI455X (gfx1250) — compile-verified
//
#include <hip/hip_runtime.h>

typedef __attribute__((ext_vector_type(2))) float v2f;
typedef __attribute__((ext_vector_type(8))) float v8f;

// One block handles one (patch n, channel c) slice:
//   t = relu(x * std + mean)          [S x S]
//   U = P @ t                          [56->64 x S]
//   Y = U @ P^T                        [56 x 56]
// using V_WMMA_F32_16X16X4_F32 for both GEMMs.
template <int S>
__global__ __launch_bounds__(128) void adain_pool_wmma(
    const float* __restrict__ x,     // [32, 256, S, S]
    const float* __restrict__ smean, // [96, 256]
    const float* __restrict__ sstd,  // [96, 256]
    float* __restrict__ out,         // [96, 256, 56, 56]
    int lo)                          // patch offset of this size group
{
    constexpr int NP   = (S + 15) & ~15; // t columns padded to 16 (32,32,48)
    constexpr int KT   = S / 4;          // K tiles of 4 (6,8,12)
    constexpr int NT1  = NP / 16;        // N tiles for GEMM1 (2,2,3)
    constexpr int TSTR = NP + 1;         // odd strides -> no 4-way LDS conflicts
    constexpr int PSTR = S + 1;
    constexpr int USTR = NP + 1;

    __shared__ float tP[64 * PSTR]; // P, zero-padded to 64 rows
    __shared__ float tT[S * TSTR];  // relu(x*std+mean), cols zero-padded to NP
    __shared__ float tU[64 * USTR]; // U = P @ t

    const int bx  = blockIdx.x;
    const int n   = bx >> 8;
    const int c   = bx & 255;
    const int tid = threadIdx.x;

    const float mu = smean[(lo + n) * 256 + c];
    const float sd = sstd [(lo + n) * 256 + c];

    // Zero LDS (covers zero-padding of t columns and P rows 56..63).
    for (int i = tid; i < S * TSTR; i += 128) tT[i] = 0.0f;
    for (int i = tid; i < 64 * PSTR; i += 128) tP[i] = 0.0f;
    __syncthreads();

    // Stage normalized+ReLU'd input.
    const float* xs = x + (size_t)(n * 256 + c) * (S * S);
    for (int i = tid; i < S * S; i += 128) {
        int r = i / S, col = i - r * S;
        float v = fmaf(xs[i], sd, mu);
        tT[r * TSTR + col] = v > 0.0f ? v : 0.0f;
    }
    // Exact AdaptiveAvgPool matrix: P[i][j] = 1/(end-start) on [start,end).
    for (int i = tid; i < 56 * S; i += 128) {
        int row = i / S, col = i - row * S;
        int st = (row * S) / 56;
        int en = ((row + 1) * S + 55) / 56; // ceil
        tP[row * PSTR + col] = (col >= st && col < en) ? 1.0f / (float)(en - st) : 0.0f;
    }
    __syncthreads();

    const int wave  = tid >> 5;
    const int lane  = tid & 31;
    const int m0    = wave * 16;        // 16-row band per wave (M = 64)
    const int lrow  = lane & 15;        // intra-tile index along lanes
    const int khalf = (lane >> 4) * 2;  // fp32 A/B: K = r + 2*(lane>=16)
    const int rhalf = (lane >> 4) * 8;  // fp32 C/D: M = r + 8*(lane>=16)

    // A tiles of GEMM1 (rows of P) are N-invariant: hoist into registers.
    v2f aP[KT];
#pragma unroll
    for (int kt = 0; kt < KT; ++kt) {
        const int kb = kt * 4 + khalf;
        aP[kt].x = tP[(m0 + lrow) * PSTR + kb + 0];
        aP[kt].y = tP[(m0 + lrow) * PSTR + kb + 1];
    }

    // ---- GEMM1: U[64 x NP] = P @ t ----
    for (int nt = 0; nt < NT1; ++nt) {
        v8f acc = {};
#pragma unroll
        for (int kt = 0; kt < KT; ++kt) {
            const int kb = kt * 4 + khalf;
            v2f b;
            b.x = tT[(kb + 0) * TSTR + nt * 16 + lrow];
            b.y = tT[(kb + 1) * TSTR + nt * 16 + lrow];
            acc = __builtin_amdgcn_wmma_f32_16x16x4_f32(
                false, aP[kt], false, b, (short)0, acc, false, false);
        }
#pragma unroll
        for (int r = 0; r < 8; ++r)
            tU[(m0 + r + rhalf) * USTR + nt * 16 + lrow] = acc[r];
    }
    __syncthreads();

    // A tiles of GEMM2 (rows of U), also N-invariant.
    v2f aU[KT];
#pragma unroll
    for (int kt = 0; kt < KT; ++kt) {
        const int kb = kt * 4 + khalf;
        aU[kt].x = tU[(m0 + lrow) * USTR + kb + 0];
        aU[kt].y = tU[(m0 + lrow) * USTR + kb + 1];
    }

    // ---- GEMM2: Y[56 x 56] = U @ P^T ----
    const size_t ob = ((size_t)(lo + n) * 256 + c) * (56 * 56);
    for (int nt = 0; nt < 4; ++nt) {
        v8f acc = {};
#pragma unroll
        for (int kt = 0; kt < KT; ++kt) {
            const int kb = kt * 4 + khalf;
            v2f b; // P^T[k][n] = P[n][k]
            b.x = tP[(nt * 16 + lrow) * PSTR + kb + 0];
            b.y = tP[(nt * 16 + lrow) * PSTR + kb + 1];
            acc = __builtin_amdgcn_wmma_f32_16x16x4_f32(
                false, aU[kt], false, b, (short)0, acc, false, false);
        }
        const int col = nt * 16 + lrow;
#pragma unroll
        for (int r = 0; r < 8; ++r) {
            const int row = m0 + r + rhalf;
            if (row < 56 && col < 56)
                out[ob + row * 56 + col] = acc[r];
        }
    }
}

extern "C" void kernel_launch(void* const* d_in, const int* in_sizes, int n_in,
                              void* d_out, int out_size, void* d_ws, size_t ws_size,
                              hipStream_t stream) {
    (void)in_sizes; (void)n_in; (void)out_size; (void)d_ws; (void)ws_size;
    const float* p24 = (const float*)d_in[0]; // [32,256,24,24]
    const float* p32 = (const float*)d_in[1]; // [32,256,32,32]
    const float* p48 = (const float*)d_in[2]; // [32,256,48,48]
    const float* sm  = (const float*)d_in[3]; // [96,256,1,1]
    const float* ss  = (const float*)d_in[4]; // [96,256,1,1]
    float* out = (float*)d_out;               // [96,256,56,56]

    dim3 grid(32 * 256);
    dim3 block(128);
    adain_pool_wmma<24><<<grid, block, 0, stream>>>(p24, sm, ss, out, 0);
    adain_pool_wmma<32><<<grid, block, 0, stream>>>(p32, sm, ss, out, 32);
    adain_pool_wmma<48><<<grid, block, 0, stream>>>(p48, sm, ss, out, 64);
}